// LearnedEquivariance_57114475102852
// MI455X (gfx1250) — compile-verified
//
#include <hip/hip_runtime.h>
#include <stdint.h>

// Problem constants (from the reference)
#define HH 128
#define WW 128
#define KK 5
#define GG 40
#define CC 32

typedef __attribute__((ext_vector_type(4))) unsigned int u32x4;
typedef __attribute__((ext_vector_type(8))) unsigned int u32x8;

// TDM D# group 0: count=1 (valid user descriptor), lds byte address,
// 57-bit global byte address, type=2 ("image").
static __device__ __forceinline__ u32x4 tdm_group0(uint32_t lds_addr, uint64_t gaddr) {
    u32x4 g;
    g.x = 1u;
    g.y = lds_addr;
    g.z = (uint32_t)(gaddr & 0xFFFFFFFFull);
    g.w = (uint32_t)((gaddr >> 32) & 0x01FFFFFFull) | (2u << 30);
    return g;
}

// TDM D# group 1 for a dense 2D 128x128 tile of 4-byte elements:
// data_size=2 (4B), tensor_dim0=128, tensor_dim1=128, tile_dim0=128,
// tile_dim1=128, tensor_dim0_stride=128, tensor_dim1_stride=128*128.
// No multicast / pad / iterate / atomic-barrier.
static __device__ __forceinline__ u32x8 tdm_group1_plane() {
    u32x8 g;
    g[0] = (2u << 16);
    g[1] = ((uint32_t)WW) << 16;                     // tensor_dim0 lo16
    g[2] = ((uint32_t)HH) << 16;                     // dim0 hi16(0) | tensor_dim1 lo16
    g[3] = ((uint32_t)WW) << 16;                     // dim1 hi16(0) | tile_dim0
    g[4] = (uint32_t)HH;                             // tile_dim1 | tile_dim2(0)
    g[5] = (uint32_t)WW;                             // tensor_dim0_stride lo32
    g[6] = (((uint32_t)(HH * WW)) & 0xFFFFu) << 16;  // stride0 hi16(0) | stride1 lo16
    g[7] = ((uint32_t)(HH * WW)) >> 16;              // stride1 bits 47:16 (=0)
    return g;
}

__global__ __launch_bounds__(256)
void LearnedEquivariance_57114475102852_kernel(const float* __restrict__ x,
                                               const float* __restrict__ kernels,
                                               const int* __restrict__ g,
                                               const int* __restrict__ n_ptr,
                                               float* __restrict__ out)
{
    // Two ping-pong plane buffers: 2 * 64 KB = 128 KB LDS (320 KB/WGP -> 2 WG/WGP)
    __shared__ float buf0[HH * WW];
    __shared__ float buf1[HH * WW];

    const int plane = blockIdx.x;       // = b*CC + c
    const int b = plane >> 5;           // CC == 32

    // Per-sample depthwise 5x5 weights (block-uniform) -> SGPRs
    int grp = g[b] % GG; if (grp < 0) grp += GG;
    const float* __restrict__ wp = kernels + (size_t)grp * (KK * KK);
    float wreg[KK * KK];
    #pragma unroll
    for (int i = 0; i < KK * KK; ++i) wreg[i] = wp[i];

    const int n = n_ptr[0];             // number of conv applications (4)

    const uint32_t lds0 = (uint32_t)(uintptr_t)(void*)buf0;
    const uint32_t lds1 = (uint32_t)(uintptr_t)(void*)buf1;

    // ---- TDM: DMA the whole 128x128 fp32 plane HBM -> LDS (one descriptor) ----
    if (threadIdx.x == 0) {
        const uint64_t src_gaddr = (uint64_t)(uintptr_t)(x + (size_t)plane * (HH * WW));
        u32x4 g0 = tdm_group0(lds0, src_gaddr);
        u32x8 g1 = tdm_group1_plane();
        asm volatile("tensor_load_to_lds %0, %1" :: "s"(g0), "s"(g1) : "memory");
        __builtin_amdgcn_s_wait_tensorcnt(0);
    }
    __syncthreads();

    // Thread -> (x pair, row quarter). 4-row x 2-col register tile, 8 tiles/pass.
    // 200 FMA per 48 LDS loads (~4.2 FMA/word): well under the 64-bank LDS rate.
    const int x0    = (threadIdx.x & 63) << 1;       // even column base
    const int ybase = (threadIdx.x >> 6) << 5;       // 0, 32, 64, 96
    int xc[KK + 1];
    #pragma unroll
    for (int j = 0; j < KK + 1; ++j) xc[j] = (x0 + j - 2) & (WW - 1);

    float* src = buf0;
    float* dst = buf1;
    for (int it = 0; it < n; ++it) {
        #pragma unroll 1
        for (int t = 0; t < 8; ++t) {
            const int y0 = ybase + t * 4;
            float acc[4][2] = {{0.f,0.f},{0.f,0.f},{0.f,0.f},{0.f,0.f}};
            #pragma unroll
            for (int rr = 0; rr < 8; ++rr) {          // 8 input rows feed 4 output rows
                const float* rp = src + (((y0 + rr - 2) & (HH - 1)) << 7);
                float v[KK + 1];
                #pragma unroll
                for (int j = 0; j < KK + 1; ++j) v[j] = rp[xc[j]];  // stride-2 lanes: conflict-free
                #pragma unroll
                for (int oo = 0; oo < 4; ++oo) {
                    const int i = rr - oo;
                    if (i >= 0 && i < KK) {
                        #pragma unroll
                        for (int j = 0; j < KK; ++j) {
                            const float w = wreg[i * KK + j];       // SGPR operand
                            acc[oo][0] = fmaf(w, v[j],     acc[oo][0]);
                            acc[oo][1] = fmaf(w, v[j + 1], acc[oo][1]);
                        }
                    }
                }
            }
            #pragma unroll
            for (int oo = 0; oo < 4; ++oo) {
                float2 o; o.x = acc[oo][0]; o.y = acc[oo][1];
                *(float2*)(dst + (((y0 + oo) << 7) + x0)) = o;       // ds_store_b64
            }
        }
        __syncthreads();                              // publish pass, then swap
        float* tmp = src; src = dst; dst = tmp;
    }

    // ---- TDM: DMA result plane LDS -> HBM ----
    if (threadIdx.x == 0) {
        const uint64_t dst_gaddr = (uint64_t)(uintptr_t)(out + (size_t)plane * (HH * WW));
        const uint32_t lds_res = (src == buf0) ? lds0 : lds1;
        u32x4 g0 = tdm_group0(lds_res, dst_gaddr);
        u32x8 g1 = tdm_group1_plane();
        asm volatile("tensor_store_from_lds %0, %1" :: "s"(g0), "s"(g1) : "memory");
        __builtin_amdgcn_s_wait_tensorcnt(0);
    }
}

extern "C" void kernel_launch(void* const* d_in, const int* in_sizes, int n_in,
                              void* d_out, int out_size, void* d_ws, size_t ws_size,
                              hipStream_t stream) {
    const float* x       = (const float*)d_in[0];   // [B, 32, 128, 128] fp32
    const float* kernels = (const float*)d_in[1];   // [40, 5, 5] fp32
    const int*   g       = (const int*)d_in[2];     // [B] int
    // d_in[3] = l (unused on the num_layers==0 path)
    const int*   n_ptr   = (const int*)d_in[4];     // scalar n (=4)
    float*       out     = (float*)d_out;

    const int B = in_sizes[2];                      // g has B elements
    dim3 grid(B * CC);
    dim3 block(256);
    hipLaunchKernelGGL(LearnedEquivariance_57114475102852_kernel,
                       grid, block, 0, stream, x, kernels, g, n_ptr, out);
}